// Expansion_and_CPSDropout_74801150427237
// MI455X (gfx1250) — compile-verified
//
#include <hip/hip_runtime.h>
#include <cstdint>

// Expansion (5x nearest-window gather) + correlated-pattern dropout, fp32.
// Store-bandwidth bound (419 MB out vs 17 MB in): stage each 16KB input
// channel into LDS once via the Tensor Data Mover, gather from LDS, emit
// non-temporal b128 stores.

typedef unsigned int u32x4 __attribute__((ext_vector_type(4)));
typedef int          i32x8 __attribute__((ext_vector_type(8)));
typedef int          i32x4 __attribute__((ext_vector_type(4)));
typedef float        f32x4 __attribute__((ext_vector_type(4)));

#define RATE     0.25f
#define INV_KEEP (1.0f / 0.75f)

__global__ __launch_bounds__(256)
void expand_cps_dropout_kernel(const float* __restrict__ x,          // (4,256,64,64)
                               const float* __restrict__ rand_vals,  // (320,320)
                               float* __restrict__ out)              // (4,256,320,320)
{
    __shared__ float tileX[64 * 64];   // 16 KB: one full input channel

    const int ch     = blockIdx.x;     // 0..1023  (b*c flattened)
    const int rowblk = blockIdx.y;     // 0..4     (64 output rows each)
    const int tid    = threadIdx.x;

    // ---- TDM: DMA the 64x64 fp32 channel (4096 dwords) global -> LDS ----
    if (tid == 0) {
        const unsigned long long ga =
            (unsigned long long)(uintptr_t)(x + (size_t)ch * 4096);
        const unsigned int lds = (unsigned int)(uintptr_t)(void*)tileX;

        u32x4 g0;
        g0.x = 0x1u;                                   // count=1, user desc, no gather
        g0.y = lds;                                    // lds_addr
        g0.z = (unsigned int)(ga & 0xFFFFFFFFull);     // global_addr[31:0]
        g0.w = (unsigned int)((ga >> 32) & 0x01FFFFFFull) | (2u << 30); // addr hi | type=2

        i32x8 g1;
        g1[0] = 0x00020000;                  // workgroup_mask=0, data_size=2 (4B)
        g1[1] = (int)(4096u << 16);          // tensor_dim0 = 4096 (lo 16 bits)
        g1[2] = (int)(1u << 16);             // tensor_dim0 hi=0, tensor_dim1 = 1
        g1[3] = (int)(4096u << 16);          // tile_dim0 = 4096
        g1[4] = 0;                           // tile_dim1/2 unused
        g1[5] = 4096;                        // tensor_dim0_stride
        g1[6] = 0;
        g1[7] = 0;

        i32x4 z4 = {0, 0, 0, 0};             // groups 2/3 unused (<=2D tensor)
        i32x8 z8 = {0, 0, 0, 0, 0, 0, 0, 0};
        __builtin_amdgcn_tensor_load_to_lds(g0, g1, z4, z4, z8, 0);
        __builtin_amdgcn_s_wait_tensorcnt(0);
    }
    __syncthreads();

    // ---- Expand + mask: 64 rows x 320 cols per block = 5120 float4 stores ----
    const f32x4* __restrict__ rnd4 = (const f32x4*)rand_vals;
    f32x4* __restrict__ out4 = (f32x4*)out + (size_t)ch * 25600;

    for (int k = 0; k < 20; ++k) {
        const int fid = rowblk * 5120 + k * 256 + tid;  // float4 index in channel
        const int r   = fid / 80;                        // output row 0..319
        const int g   = fid - r * 80;                    // float4 column 0..79

        const int rq = r / 5;
        const int rm = r - rq * 5;
        const int ih = rq + rm - 2;                      // padded-gather input row
        const bool rowok = ((unsigned)ih < 64u);

        const f32x4 rv = rnd4[fid];                      // mask values, same layout
        const int ow0 = g * 4;

        f32x4 o;
        #pragma unroll
        for (int j = 0; j < 4; ++j) {
            const int ow = ow0 + j;
            const int wq = ow / 5;
            const int wm = ow - wq * 5;
            const int iw = wq + wm - 2;                  // padded-gather input col
            const float v = (rowok && (unsigned)iw < 64u) ? tileX[ih * 64 + iw] : 0.0f;
            const bool keep = (rv[j] > RATE) || ((rm == 2) & (wm == 2));
            o[j] = keep ? v * INV_KEEP : 0.0f;
        }
        // Streaming 419 MB output: write-once, bypass cache residency.
        __builtin_nontemporal_store(o, &out4[fid]);
    }
}

extern "C" void kernel_launch(void* const* d_in, const int* in_sizes, int n_in,
                              void* d_out, int out_size, void* d_ws, size_t ws_size,
                              hipStream_t stream) {
    const float* x         = (const float*)d_in[0];  // (4,256,64,64) f32
    const float* rand_vals = (const float*)d_in[1];  // (320,320)     f32
    float*       out       = (float*)d_out;          // (4,256,320,320) f32

    dim3 grid(1024, 5);   // 1024 (b*c) channels x 5 row-blocks of 64 rows
    expand_cps_dropout_kernel<<<grid, 256, 0, stream>>>(x, rand_vals, out);
}